// MultiSourceGNNBlock_1039382085724
// MI455X (gfx1250) — compile-verified
//
#include <hip/hip_runtime.h>

typedef __attribute__((ext_vector_type(2))) float v2f;
typedef __attribute__((ext_vector_type(8))) float v8f;

#define HEADS 4
#define DHEAD 64
#define FDIM  256            // HEADS * DHEAD
#define NEG_SLOPE 0.2f

// ---------------------------------------------------------------------------
// GEMM: Y[M,FDIM] = X[M,K] @ W[K,FDIM]   (M multiple of 16, K multiple of 4)
// 256 threads (8 waves) per block; block owns 16 rows, all 256 cols.
// Uses V_WMMA_F32_16X16X4_F32 (full f32 precision, ISA 7.12 layouts).
// ---------------------------------------------------------------------------
__global__ __launch_bounds__(256) void gemm16_wmma_f32(
    const float* __restrict__ X, const float* __restrict__ W,
    float* __restrict__ Y, int K) {
  __shared__ float Xs[16 * 256];                 // 16KB, covers K <= 256
  const int tid = threadIdx.x;
  const int m0  = blockIdx.x * 16;

  for (int i = tid; i < 16 * K; i += 256)
    Xs[i] = X[(size_t)(m0 + i / K) * K + (i % K)];
  __syncthreads();

  const int wave = tid >> 5;                     // 0..7
  const int lane = tid & 31;
  const int half = lane >> 4;                    // 0 or 1
  const int l16  = lane & 15;
  const int n0   = wave * 32;                    // two 16-col tiles per wave

  v8f c0 = {0.f, 0.f, 0.f, 0.f, 0.f, 0.f, 0.f, 0.f};
  v8f c1 = {0.f, 0.f, 0.f, 0.f, 0.f, 0.f, 0.f, 0.f};

  for (int kk = 0; kk < K; kk += 4) {
    const int ka = kk + 2 * half;
    v2f a, b0, b1;
    a[0]  = Xs[l16 * K + ka];
    a[1]  = Xs[l16 * K + ka + 1];
    b0[0] = W[(size_t)(ka    ) * FDIM + n0 + l16];
    b0[1] = W[(size_t)(ka + 1) * FDIM + n0 + l16];
    b1[0] = W[(size_t)(ka    ) * FDIM + n0 + 16 + l16];
    b1[1] = W[(size_t)(ka + 1) * FDIM + n0 + 16 + l16];
    c0 = __builtin_amdgcn_wmma_f32_16x16x4_f32(false, a, false, b0,
                                               (short)0, c0, false, false);
    c1 = __builtin_amdgcn_wmma_f32_16x16x4_f32(false, a, false, b1,
                                               (short)0, c1, false, false);
  }

#pragma unroll
  for (int r = 0; r < 8; ++r) {
    const size_t row = (size_t)(m0 + r + 8 * half);
    Y[row * FDIM + n0 + l16]      = c0[r];
    Y[row * FDIM + n0 + 16 + l16] = c1[r];
  }
}

// ---------------------------------------------------------------------------
// Per-(node, head) attention coefficients: one wave32 per task.
// alpha_s[n,h] = sum_d xh[n,h,d]*a_src[h,d] ; same for alpha_d.
// ---------------------------------------------------------------------------
__global__ __launch_bounds__(256) void alpha_kernel(
    const float* __restrict__ xh, const float* __restrict__ a_s,
    const float* __restrict__ a_d, float* __restrict__ alpha_s,
    float* __restrict__ alpha_d, int BN) {
  const int task = blockIdx.x * 8 + (threadIdx.x >> 5);
  if (task >= BN * HEADS) return;
  const int lane = threadIdx.x & 31;
  const int n = task >> 2, h = task & 3;
  const size_t base = (size_t)n * FDIM + h * DHEAD;
  const float x0 = xh[base + lane];
  const float x1 = xh[base + lane + 32];
  float ss = x0 * a_s[h * DHEAD + lane] + x1 * a_s[h * DHEAD + lane + 32];
  float sd = x0 * a_d[h * DHEAD + lane] + x1 * a_d[h * DHEAD + lane + 32];
#pragma unroll
  for (int off = 16; off > 0; off >>= 1) {
    ss += __shfl_xor(ss, off, 32);
    sd += __shfl_xor(sd, off, 32);
  }
  if (lane == 0) { alpha_s[task] = ss; alpha_d[task] = sd; }
}

__global__ void fill_kernel(float* __restrict__ p, float v, int n) {
  const int i = blockIdx.x * blockDim.x + threadIdx.x;
  if (i < n) p[i] = v;
}

// Sign-aware float atomic max (monotone int/uint bit ordering).
__device__ __forceinline__ void atomicMaxF(float* addr, float v) {
  if (v >= 0.f) atomicMax((int*)addr, __float_as_int(v));
  else          atomicMin((unsigned int*)addr, __float_as_uint(v));
}

// ---------------------------------------------------------------------------
// Edge logits + running segment max. Edges [0,E) from edge_index (row-major
// [2,E]); edges [E, E+BN) are the implicit self loops.
// ---------------------------------------------------------------------------
__global__ void logits_kernel(const int* __restrict__ ei, int E, int BN,
                              const float* __restrict__ alpha_s,
                              const float* __restrict__ alpha_d,
                              float* __restrict__ ebuf,
                              float* __restrict__ emax) {
  const int t = blockIdx.x * blockDim.x + threadIdx.x;
  const int Et = E + BN;
  if (t >= Et * HEADS) return;
  const int e = t >> 2, h = t & 3;
  int src, dst;
  if (e < E) { src = ei[e]; dst = ei[E + e]; }
  else       { src = dst = e - E; }
  float v = alpha_s[src * HEADS + h] + alpha_d[dst * HEADS + h];
  v = (v > 0.f) ? v : NEG_SLOPE * v;
  ebuf[t] = v;
  atomicMaxF(&emax[dst * HEADS + h], v);
}

__global__ void expden_kernel(const int* __restrict__ ei, int E, int BN,
                              float* __restrict__ ebuf,
                              const float* __restrict__ emax,
                              float* __restrict__ den) {
  const int t = blockIdx.x * blockDim.x + threadIdx.x;
  const int Et = E + BN;
  if (t >= Et * HEADS) return;
  const int e = t >> 2, h = t & 3;
  const int dst = (e < E) ? ei[E + e] : (e - E);
  const float v = __expf(ebuf[t] - emax[dst * HEADS + h]);
  ebuf[t] = v;
  atomicAdd(&den[dst * HEADS + h], v);
}

// One wave32 per (edge, head): 64-wide weighted message scatter-add.
__global__ __launch_bounds__(256) void scatter_kernel(
    const int* __restrict__ ei, int E, int BN,
    const float* __restrict__ xh, const float* __restrict__ ebuf,
    const float* __restrict__ den, float* __restrict__ acc) {
  const int wid = blockIdx.x * 8 + (threadIdx.x >> 5);
  const int Et = E + BN;
  if (wid >= Et * HEADS) return;
  const int lane = threadIdx.x & 31;
  const int e = wid >> 2, h = wid & 3;
  int src, dst;
  if (e < E) { src = ei[e]; dst = ei[E + e]; }
  else       { src = dst = e - E; }
  const float att = ebuf[wid] / (den[dst * HEADS + h] + 1e-16f);
  const float* xr = xh  + (size_t)src * FDIM + h * DHEAD;
  float*      orow = acc + (size_t)dst * FDIM + h * DHEAD;
  atomicAdd(&orow[lane],      xr[lane]      * att);
  atomicAdd(&orow[lane + 32], xr[lane + 32] * att);
}

__global__ void biasact_kernel(const float* __restrict__ acc,
                               const float* __restrict__ bias,
                               float* __restrict__ out, int BN, int do_relu) {
  const int i = blockIdx.x * blockDim.x + threadIdx.x;
  if (i >= BN * FDIM) return;
  float v = acc[i] + bias[i & (FDIM - 1)];
  out[i] = (do_relu && v < 0.f) ? 0.f : v;
}

// ---------------------------------------------------------------------------
static void run_gat_layer(const float* X, int K, const int* ei, int E, int BN,
                          const float* W, const float* a_s, const float* a_d,
                          const float* bias, float* xh, float* acc, float* outp,
                          int do_relu, float* alpha_s, float* alpha_d,
                          float* emax, float* den, float* ebuf,
                          hipStream_t stream) {
  const int Et = E + BN;
  gemm16_wmma_f32<<<BN / 16, 256, 0, stream>>>(X, W, xh, K);
  alpha_kernel<<<(BN * HEADS + 7) / 8, 256, 0, stream>>>(xh, a_s, a_d,
                                                         alpha_s, alpha_d, BN);
  fill_kernel<<<(BN * FDIM + 255) / 256, 256, 0, stream>>>(acc, 0.f, BN * FDIM);
  fill_kernel<<<(BN * HEADS + 255) / 256, 256, 0, stream>>>(emax, -1e30f, BN * HEADS);
  fill_kernel<<<(BN * HEADS + 255) / 256, 256, 0, stream>>>(den, 0.f, BN * HEADS);
  logits_kernel<<<(Et * HEADS + 255) / 256, 256, 0, stream>>>(ei, E, BN,
                                                              alpha_s, alpha_d,
                                                              ebuf, emax);
  expden_kernel<<<(Et * HEADS + 255) / 256, 256, 0, stream>>>(ei, E, BN,
                                                              ebuf, emax, den);
  scatter_kernel<<<(Et * HEADS + 7) / 8, 256, 0, stream>>>(ei, E, BN,
                                                           xh, ebuf, den, acc);
  biasact_kernel<<<(BN * FDIM + 255) / 256, 256, 0, stream>>>(acc, bias, outp,
                                                              BN, do_relu);
}

extern "C" void kernel_launch(void* const* d_in, const int* in_sizes, int n_in,
                              void* d_out, int out_size, void* d_ws, size_t ws_size,
                              hipStream_t stream) {
  const float* h0 = (const float*)d_in[0];
  const int* eis[2] = {(const int*)d_in[1], (const int*)d_in[2]};
  // params: per source s, per layer l
  const float* W [2][2] = {{(const float*)d_in[3],  (const float*)d_in[7]},
                           {(const float*)d_in[11], (const float*)d_in[15]}};
  const float* As[2][2] = {{(const float*)d_in[4],  (const float*)d_in[8]},
                           {(const float*)d_in[12], (const float*)d_in[16]}};
  const float* Ad[2][2] = {{(const float*)d_in[5],  (const float*)d_in[9]},
                           {(const float*)d_in[13], (const float*)d_in[17]}};
  const float* Bi[2][2] = {{(const float*)d_in[6],  (const float*)d_in[10]},
                           {(const float*)d_in[14], (const float*)d_in[18]}};

  const int BN = in_sizes[0] / DHEAD;   // 8000 nodes total (B*N)
  const int E  = in_sizes[1] / 2;       // 256000 edges per source
  const int Et = E + BN;

  float* ws = (float*)d_ws;
  float* xh      = ws; ws += (size_t)BN * FDIM;
  float* acc     = ws; ws += (size_t)BN * FDIM;
  float* hmid    = ws; ws += (size_t)BN * FDIM;
  float* alpha_s = ws; ws += (size_t)BN * HEADS;
  float* alpha_d = ws; ws += (size_t)BN * HEADS;
  float* emax    = ws; ws += (size_t)BN * HEADS;
  float* den     = ws; ws += (size_t)BN * HEADS;
  float* ebuf    = ws; ws += (size_t)Et * HEADS;

  for (int s = 0; s < 2; ++s) {
    float* outp = (float*)d_out + (size_t)s * BN * FDIM;
    // layer 0: h0 [BN,64] -> hmid [BN,256], ReLU
    run_gat_layer(h0, DHEAD, eis[s], E, BN, W[s][0], As[s][0], Ad[s][0],
                  Bi[s][0], xh, acc, hmid, 1,
                  alpha_s, alpha_d, emax, den, ebuf, stream);
    // layer 1: hmid [BN,256] -> outp [BN,256], no activation
    run_gat_layer(hmid, FDIM, eis[s], E, BN, W[s][1], As[s][1], Ad[s][1],
                  Bi[s][1], xh, acc, outp, 0,
                  alpha_s, alpha_d, emax, den, ebuf, stream);
  }
}